// SoftBinHaversineLoss_4750233829711
// MI455X (gfx1250) — compile-verified
//
#include <hip/hip_runtime.h>
#include <hip/hip_bf16.h>

#define TAU      0.25f
#define INV_TAU  4.0f
#define EARTH_R  3958.8f
#define D2R      0.017453292519943295f

typedef __attribute__((ext_vector_type(2))) float v2f;
typedef __attribute__((ext_vector_type(8))) float v8f;

// One workgroup (8 wave32s) per prediction row. Streams the row's bin
// (4096 float2, L2-resident) with prefetch, does per-thread online
// logsumexp, block max, then an exact-f32 WMMA lane reduction.
__global__ __launch_bounds__(256) void softbin_row_kernel(
    const float* __restrict__ preds,       // (B,2) lon,lat
    const float* __restrict__ bin_coords,  // (NBINS,M,2) lon,lat
    const int*   __restrict__ x_vals,      // (B,3)
    const int*   __restrict__ bin_counts,  // (NBINS)
    float* __restrict__ row_val,
    float* __restrict__ row_flag,
    int M)
{
    const int b    = blockIdx.x;
    const int tid  = threadIdx.x;
    const int lane = tid & 31;
    const int wid  = tid >> 5;

    const int x0 = x_vals[3 * b + 0];
    const int x1 = x_vals[3 * b + 1];
    const int x2 = x_vals[3 * b + 2];
    const int bin = x0 * 25 + x1 * 5 + x2;
    int count = bin_counts[bin];
    if (count > M) count = M;

    const float lon1    = preds[2 * b + 0] * D2R;
    const float lat1    = preds[2 * b + 1] * D2R;
    const float coslat1 = __cosf(lat1);

    const float2* __restrict__ cand =
        (const float2*)(bin_coords + (size_t)bin * (size_t)M * 2u);

    const float NEGINF = -__builtin_inff();
    float m = NEGINF;   // running max of logits
    float s = 0.0f;     // running sum of exp(logit - m)

    for (int j = tid; j < count; j += 256) {
        // speculative prefetch of next chunk -> global_prefetch_b8 (OOB is dropped)
        __builtin_prefetch(&cand[j + 256], 0, 0);
        const float2 c2 = cand[j];
        const float lon2 = c2.x * D2R;
        const float lat2 = c2.y * D2R;
        const float sdlat = __sinf(0.5f * (lat2 - lat1));
        const float sdlon = __sinf(0.5f * (lon2 - lon1));
        float a = sdlat * sdlat + coslat1 * __cosf(lat2) * (sdlon * sdlon);
        a = fminf(fmaxf(a, 0.0f), 1.0f);
        const float c = 2.0f * atan2f(__builtin_sqrtf(a), __builtin_sqrtf(1.0f - a));
        const float logit = -(EARTH_R * c) * INV_TAU;

        if (logit > m) {
            s = s * __expf(m - logit) + 1.0f;  // first iter: exp(-inf)=0 -> s=1
            m = logit;
        } else {
            s += __expf(logit - m);
        }
    }

    // ---- block-wide max of m -> Mb (broadcast) ----
    __shared__ float sh[8];
    __shared__ float shM;
    float wm = m;
    #pragma unroll
    for (int off = 16; off; off >>= 1)
        wm = fmaxf(wm, __shfl_xor(wm, off, 32));
    if (lane == 0) sh[wid] = wm;
    __syncthreads();
    if (tid == 0) {
        float Mx = sh[0];
        #pragma unroll
        for (int i = 1; i < 8; ++i) Mx = fmaxf(Mx, sh[i]);
        shM = Mx;
    }
    __syncthreads();
    const float Mb = shM;

    // rescale this thread's partial sum to the common max
    const float sd = (m == NEGINF) ? 0.0f : s * __expf(m - Mb);

    // ---- exact f32 lane reduction via V_WMMA_F32_16X16X4_F32 ----
    // A (16x4): lane L%16 row, v0 holds sd, v1 = 0. B = ones(4x16)
    // => D[m][n] = sd[lane m] + sd[lane m+16]  (identical across n)
    v2f a_op; a_op[0] = sd;   a_op[1] = 0.0f;
    v2f b_op; b_op[0] = 1.0f; b_op[1] = 1.0f;
    v8f c_op = {};
    v8f d_op = __builtin_amdgcn_wmma_f32_16x16x4_f32(
        false, a_op, false, b_op, (short)0, c_op, false, false);
    float half8 = d_op[0] + d_op[1] + d_op[2] + d_op[3] +
                  d_op[4] + d_op[5] + d_op[6] + d_op[7];
    // lanes 0-15 hold rows 0-7 sums, lanes 16-31 hold rows 8-15 sums
    const float wave_sum = half8 + __shfl_xor(half8, 16, 32);

    if (lane == 0) sh[wid] = wave_sum;
    __syncthreads();
    if (tid == 0) {
        float S = 0.0f;
        #pragma unroll
        for (int i = 0; i < 8; ++i) S += sh[i];
        float softmin = 0.0f, flag = 0.0f;
        if (count > 0) {
            softmin = -TAU * (Mb + __logf(S));
            flag = 1.0f;
        }
        row_val[b]  = softmin;
        row_flag[b] = flag;
    }
}

// Deterministic single-block reduction (fixed-order tree; no atomics).
__global__ __launch_bounds__(256) void softbin_finalize_kernel(
    const float* __restrict__ row_val,
    const float* __restrict__ row_flag,
    int B, float* __restrict__ out)
{
    const int tid = threadIdx.x, lane = tid & 31, wid = tid >> 5;
    float s = 0.0f, c = 0.0f;
    for (int j = tid; j < B; j += 256) {
        s += row_val[j];
        c += row_flag[j];
    }
    #pragma unroll
    for (int off = 16; off; off >>= 1) {
        s += __shfl_xor(s, off, 32);
        c += __shfl_xor(c, off, 32);
    }
    __shared__ float shs[8], shc[8];
    if (lane == 0) { shs[wid] = s; shc[wid] = c; }
    __syncthreads();
    if (tid == 0) {
        float S = 0.0f, C = 0.0f;
        #pragma unroll
        for (int i = 0; i < 8; ++i) { S += shs[i]; C += shc[i]; }
        out[0] = S / fmaxf(C, 1.0f);
    }
}

extern "C" void kernel_launch(void* const* d_in, const int* in_sizes, int n_in,
                              void* d_out, int out_size, void* d_ws, size_t ws_size,
                              hipStream_t stream) {
    const float* preds      = (const float*)d_in[0];   // (B,2) f32
    const float* bin_coords = (const float*)d_in[1];   // (NBINS,M,2) f32
    const int*   x_vals     = (const int*)d_in[2];     // (B,3) int
    const int*   bin_counts = (const int*)d_in[3];     // (NBINS,) int

    const int B     = in_sizes[0] / 2;                 // 8192
    const int NBINS = in_sizes[3];                     // 125
    const int M     = in_sizes[1] / (2 * NBINS);       // 4096

    float* row_val  = (float*)d_ws;                    // B floats
    float* row_flag = row_val + B;                     // B floats

    softbin_row_kernel<<<B, 256, 0, stream>>>(
        preds, bin_coords, x_vals, bin_counts, row_val, row_flag, M);
    softbin_finalize_kernel<<<1, 256, 0, stream>>>(
        row_val, row_flag, B, (float*)d_out);
}